// Attack_PGD_86260123173299
// MI455X (gfx1250) — compile-verified
//
#include <hip/hip_runtime.h>
#include <math.h>

#define BB   128
#define CC   3
#define HH   224
#define PN   14
#define PS   16
#define NP   196
#define TOPK 20
#define NCLS 1000
#define KDIM 588     // C*NP
#define NPAD 1008    // 63*16

typedef __attribute__((ext_vector_type(2))) float v2f;
typedef __attribute__((ext_vector_type(8))) float v8f;

#define NEG_INF (-3.402823466e38f)

// ---------------------------------------------------------------------------
// Kernel 1: partial sums of patch_grad over (batch, dim) per patch.
// grid (196, 4), block 256 (192 active loaders). float4 loads: 768 = 192*4.
__global__ void k_patch_sum(const float* __restrict__ pg, float* __restrict__ partial) {
    int p = blockIdx.x, s = blockIdx.y, t = threadIdx.x;
    float acc = 0.f;
    const float4* pg4 = (const float4*)pg;
    if (t < 192) {
        int b0 = s * 32;
        for (int b = b0; b < b0 + 32; ++b) {
            float4 v = pg4[((size_t)b * NP + p) * 192 + t];
            acc += (v.x + v.y) + (v.z + v.w);
        }
    }
    __shared__ float red[256];
    red[t] = acc; __syncthreads();
    for (int off = 128; off > 0; off >>= 1) {
        if (t < off) red[t] += red[t + off];
        __syncthreads();
    }
    if (t == 0) partial[p * 4 + s] = red[0];
}

// ---------------------------------------------------------------------------
// Kernel 2: deterministic top-20 of 196 sums -> 0/1 selection mask sel[196].
__global__ void k_topk(const float* __restrict__ partial, float* __restrict__ sel) {
    __shared__ float vals[NP];
    __shared__ float sel_s[NP];
    __shared__ float rv[256];
    __shared__ int   ri[256];
    int t = threadIdx.x;
    if (t < NP) {
        vals[t] = partial[t * 4] + partial[t * 4 + 1] + partial[t * 4 + 2] + partial[t * 4 + 3];
        sel_s[t] = 0.f;
    }
    __syncthreads();
    for (int it = 0; it < TOPK; ++it) {
        rv[t] = (t < NP) ? vals[t] : NEG_INF;
        ri[t] = t;
        __syncthreads();
        for (int off = 128; off > 0; off >>= 1) {
            if (t < off) {
                float vo = rv[t + off]; int io = ri[t + off];
                if (vo > rv[t] || (vo == rv[t] && io < ri[t])) { rv[t] = vo; ri[t] = io; }
            }
            __syncthreads();
        }
        if (t == 0) { sel_s[ri[0]] = 1.f; vals[ri[0]] = NEG_INF; }
        __syncthreads();
    }
    if (t < NP) sel[t] = sel_s[t];
}

// ---------------------------------------------------------------------------
// Pooling core: one block per image plane [224x224]. 224 threads:
//   col4 = t % 56  (float4 column: pixel cols 4*col4 .. 4*col4+3)
//   rgrp = t / 4? no: t / 56 in 0..3 (4 rows within the patch row)
// Each thread accumulates a 4x4 pixel footprint per patch row (b128 loads),
// keeps 14 partials in registers, then one LDS pass combines 16 partials
// per patch. All loads fully coalesced global_load_b128.
__device__ __forceinline__ void pool_plane(const float* __restrict__ plane,
                                           float* __restrict__ out196,
                                           float partials[PN], int t) {
    int col4 = t % 56, rgrp = t / 56;
    const float4* img = (const float4*)plane;   // row stride = 56 float4
    #pragma unroll
    for (int px = 0; px < PN; ++px) {
        float acc = 0.f;
        int rbase = px * PS + rgrp * 4;
        #pragma unroll
        for (int r = 0; r < 4; ++r) {
            float4 v = img[(rbase + r) * 56 + col4];
            acc += (v.x + v.y) + (v.z + v.w);
        }
        partials[px] = acc;
    }
    (void)out196;
}

// Kernel 3: pool attack_w per patch: apool[c*196 + p]. grid 3, block 224.
__global__ void k_attack_pool(const float* __restrict__ aw, float* __restrict__ apool) {
    int c = blockIdx.x, t = threadIdx.x;
    float part[PN];
    pool_plane(aw + (size_t)c * (HH * HH), nullptr, part, t);
    __shared__ float lds[PN * 224];
    #pragma unroll
    for (int px = 0; px < PN; ++px) lds[px * 224 + t] = part[px];
    __syncthreads();
    if (t < NP) {
        int px = t / PN, py = t % PN;
        float s = 0.f;
        #pragma unroll
        for (int rg = 0; rg < 4; ++rg)
            #pragma unroll
            for (int cq = 0; cq < 4; ++cq)
                s += lds[px * 224 + rg * 56 + py * 4 + cq];
        apool[c * NP + t] = s * (1.f / 256.f);
    }
}

// Kernel 4: pooled[b, c*196 + p] = mean(inputs block) + sel[p]*apool[c,p].
// grid (128, 3), block 224.
__global__ void k_input_pool(const float* __restrict__ in, const float* __restrict__ sel,
                             const float* __restrict__ apool, float* __restrict__ pooled) {
    int b = blockIdx.x, c = blockIdx.y, t = threadIdx.x;
    float part[PN];
    pool_plane(in + ((size_t)b * CC + c) * (HH * HH), nullptr, part, t);
    __shared__ float lds[PN * 224];
    #pragma unroll
    for (int px = 0; px < PN; ++px) lds[px * 224 + t] = part[px];
    __syncthreads();
    if (t < NP) {
        int px = t / PN, py = t % PN;
        float s = 0.f;
        #pragma unroll
        for (int rg = 0; rg < 4; ++rg)
            #pragma unroll
            for (int cq = 0; cq < 4; ++cq)
                s += lds[px * 224 + rg * 56 + py * 4 + cq];
        pooled[(size_t)b * KDIM + c * NP + t] =
            s * (1.f / 256.f) + sel[t] * apool[c * NP + t];
    }
}

// ---------------------------------------------------------------------------
// Kernel 5a: transpose + zero-pad w_cls [588,1000] -> Wt [1008, 588] (N-major).
// grid (32, 19), block (32, 32). Tiled LDS transpose, coalesced both sides.
__global__ void k_wt(const float* __restrict__ W, float* __restrict__ Wt) {
    __shared__ float tile[32][33];
    int tx = threadIdx.x, ty = threadIdx.y;
    int n0 = blockIdx.x * 32, k0 = blockIdx.y * 32;
    int k = k0 + ty, n = n0 + tx;
    tile[ty][tx] = (k < KDIM && n < NCLS) ? W[(size_t)k * NCLS + n] : 0.f;
    __syncthreads();
    int nn = n0 + ty, kk = k0 + tx;
    if (nn < NPAD && kk < KDIM) Wt[(size_t)nn * KDIM + kk] = tile[tx][ty];
}

// Kernel 5b: logits[128,1008] = pooled[128,588] @ Wt^T via V_WMMA_F32_16X16X4_F32.
// Branchless: per K-step = 2x global_load_b64 + 1x v_wmma. grid 63, block 256.
__global__ void k_gemm_t(const float* __restrict__ A, const float* __restrict__ Wt,
                         float* __restrict__ logits) {
    int lane = threadIdx.x & 31;
    int wave = threadIdx.x >> 5;
    int lm = lane & 15;
    int lh = lane >> 4;
    int m0 = wave * 16;
    int n0 = blockIdx.x * 16;

    const float* arow = A  + (size_t)(m0 + lm) * KDIM + 2 * lh;
    const float* brow = Wt + (size_t)(n0 + lm) * KDIM + 2 * lh;
    v8f acc = {};
    for (int k0 = 0; k0 < KDIM; k0 += 4) {
        v2f a, bf;
        a.x  = arow[k0];  a.y  = arow[k0 + 1];   // A: M=m0+lm, K=k0+2*lh(+1)
        bf.x = brow[k0];  bf.y = brow[k0 + 1];   // B: N=n0+lm, K=k0+2*lh(+1)
        acc = __builtin_amdgcn_wmma_f32_16x16x4_f32(
            false, a, false, bf, (short)0, acc, false, false);
    }
    #pragma unroll
    for (int v = 0; v < 8; ++v)
        logits[(size_t)(m0 + v + 8 * lh) * NPAD + n0 + lm] = acc[v];
}

// Kernel 5c: fallback guarded GEMM (if ws too small for Wt). Same math.
__global__ void k_gemm_g(const float* __restrict__ A, const float* __restrict__ W,
                         float* __restrict__ logits) {
    int lane = threadIdx.x & 31;
    int wave = threadIdx.x >> 5;
    int lm = lane & 15;
    int lh = lane >> 4;
    int m0 = wave * 16;
    int n0 = blockIdx.x * 16;
    int nb = n0 + lm;
    bool nval = nb < NCLS;

    const float* arow = A + (size_t)(m0 + lm) * KDIM + 2 * lh;
    v8f acc = {};
    for (int k0 = 0; k0 < KDIM; k0 += 4) {
        v2f a, bf;
        a.x = arow[k0]; a.y = arow[k0 + 1];
        int kb = k0 + 2 * lh;
        bf.x = nval ? W[(size_t)kb * NCLS + nb] : 0.f;
        bf.y = nval ? W[(size_t)(kb + 1) * NCLS + nb] : 0.f;
        acc = __builtin_amdgcn_wmma_f32_16x16x4_f32(
            false, a, false, bf, (short)0, acc, false, false);
    }
    #pragma unroll
    for (int v = 0; v < 8; ++v)
        logits[(size_t)(m0 + v + 8 * lh) * NPAD + n0 + lm] = acc[v];
}

// ---------------------------------------------------------------------------
// Kernel 6: per-row softmax -> probs (d_out), then log-softmax of probs -> CE.
__global__ void k_softmax(const float* __restrict__ logits, const int* __restrict__ tgts,
                          float* __restrict__ probs, float* __restrict__ ce) {
    int b = blockIdx.x, t = threadIdx.x;
    const float* row = logits + (size_t)b * NPAD;
    __shared__ float red[256];
    __shared__ float pt_s;
    float x[4];
    float mx = NEG_INF;
    #pragma unroll
    for (int j = 0; j < 4; ++j) {
        int i = t + j * 256;
        x[j] = (i < NCLS) ? row[i] : NEG_INF;
        mx = fmaxf(mx, x[j]);
    }
    red[t] = mx; __syncthreads();
    for (int off = 128; off > 0; off >>= 1) { if (t < off) red[t] = fmaxf(red[t], red[t + off]); __syncthreads(); }
    mx = red[0]; __syncthreads();

    float s = 0.f;
    #pragma unroll
    for (int j = 0; j < 4; ++j) { int i = t + j * 256; if (i < NCLS) s += expf(x[j] - mx); }
    red[t] = s; __syncthreads();
    for (int off = 128; off > 0; off >>= 1) { if (t < off) red[t] += red[t + off]; __syncthreads(); }
    s = red[0]; __syncthreads();

    float inv = 1.f / s;
    int tgt = tgts[b];
    float p[4];
    float mx2 = NEG_INF;
    #pragma unroll
    for (int j = 0; j < 4; ++j) {
        int i = t + j * 256;
        if (i < NCLS) {
            p[j] = expf(x[j] - mx) * inv;
            probs[(size_t)b * NCLS + i] = p[j];
            if (i == tgt) pt_s = p[j];
            mx2 = fmaxf(mx2, p[j]);
        } else p[j] = NEG_INF;
    }
    red[t] = mx2; __syncthreads();
    for (int off = 128; off > 0; off >>= 1) { if (t < off) red[t] = fmaxf(red[t], red[t + off]); __syncthreads(); }
    mx2 = red[0]; __syncthreads();

    float s2 = 0.f;
    #pragma unroll
    for (int j = 0; j < 4; ++j) { int i = t + j * 256; if (i < NCLS) s2 += expf(p[j] - mx2); }
    red[t] = s2; __syncthreads();
    for (int off = 128; off > 0; off >>= 1) { if (t < off) red[t] += red[t + off]; __syncthreads(); }
    s2 = red[0];
    if (t == 0) ce[b] = -((pt_s - mx2) - logf(s2));
}

// ---------------------------------------------------------------------------
// Kernel 7: loss = -mean(ce)   (u == 0 in the reference)
__global__ void k_loss(const float* __restrict__ ce, float* __restrict__ out) {
    __shared__ float red[128];
    int t = threadIdx.x;
    red[t] = ce[t]; __syncthreads();
    for (int off = 64; off > 0; off >>= 1) { if (t < off) red[t] += red[t + off]; __syncthreads(); }
    if (t == 0) out[0] = -red[0] * (1.f / 128.f);
}

// ---------------------------------------------------------------------------
extern "C" void kernel_launch(void* const* d_in, const int* in_sizes, int n_in,
                              void* d_out, int out_size, void* d_ws, size_t ws_size,
                              hipStream_t stream) {
    (void)in_sizes; (void)n_in; (void)out_size;
    const float* inputs = (const float*)d_in[0];  // [128, 3, 224, 224]
    const float* pg     = (const float*)d_in[1];  // [128, 196, 768]
    const float* aw     = (const float*)d_in[2];  // [3, 224, 224]
    const float* wcls   = (const float*)d_in[3];  // [588, 1000]
    const int*   tgts   = (const int*)d_in[4];    // [128]
    float* out = (float*)d_out;                   // [128*1000] probs + [1] loss

    float* ws      = (float*)d_ws;
    float* partial = ws;                 // 784
    float* sel     = ws + 784;           // 196
    float* apool   = ws + 980;           // 588
    float* pooled  = ws + 1568;          // 128*588 = 75264
    float* logits  = ws + 76832;         // 128*1008 = 129024
    float* ce      = ws + 205856;        // 128
    float* Wt      = ws + 205984;        // 1008*588 = 592704  (total ~3.2 MB)
    const size_t need_full = (size_t)(205984 + (size_t)NPAD * KDIM) * sizeof(float);

    k_patch_sum  <<<dim3(196, 4), 256, 0, stream>>>(pg, partial);
    k_topk       <<<1,            256, 0, stream>>>(partial, sel);
    k_attack_pool<<<3,            224, 0, stream>>>(aw, apool);
    k_input_pool <<<dim3(128, 3), 224, 0, stream>>>(inputs, sel, apool, pooled);
    if (ws_size >= need_full) {
        k_wt     <<<dim3(32, 19), dim3(32, 32), 0, stream>>>(wcls, Wt);
        k_gemm_t <<<63,           256, 0, stream>>>(pooled, Wt, logits);
    } else {
        k_gemm_g <<<63,           256, 0, stream>>>(pooled, wcls, logits);
    }
    k_softmax    <<<128,          256, 0, stream>>>(logits, tgts, out, ce);
    k_loss       <<<1,            128, 0, stream>>>(ce, out + (size_t)BB * NCLS);
}